// RGCN_52905407152975
// MI455X (gfx1250) — compile-verified
//
#include <hip/hip_runtime.h>

#define NN 50000
#define EE 600000
#define DD 128
#define RR 200
#define NBASIS 4
#define ARDIM 32
#define LLAY 2
#define GG 50
#define NSAMP 2000
#define ATTIN 320
#define LDIM (LLAY * DD)   // 256

typedef __attribute__((ext_vector_type(2))) float v2f;
typedef __attribute__((ext_vector_type(8))) float v8f;

__device__ __forceinline__ v8f wmma_f32(v2f a, v2f b, v8f c) {
  // D = A(16x4) * B(4x16) + C(16x16), fp32
  return __builtin_amdgcn_wmma_f32_16x16x4_f32(false, a, false, b, (short)0, c,
                                               false, false);
}

// Async global->LDS copy of one dword (CDNA5 GLOBAL_LOAD_ASYNC_TO_LDS_B32).
// lptr's low 32 bits are the wave-relative LDS offset (ISA 10.2 aperture).
__device__ __forceinline__ void async_ld_f32(const float* gptr, void* lptr) {
  const unsigned loff = (unsigned)(unsigned long long)lptr;
  asm volatile("global_load_async_to_lds_b32 %0, %1, off"
               :
               : "v"(loff), "v"((unsigned long long)gptr)
               : "memory");
}

__device__ __forceinline__ void wait_async_all() {
  asm volatile("s_wait_asynccnt 0" ::: "memory");
}

// ---------------------------------------------------------------------------
// Node-level transforms: for each 16-row tile of x compute x @ M for one of
// 7 matrices M (selected by blockIdx.y):
//   0..3 : weights_l[b]            -> z[:, b*128 : (b+1)*128]
//   4    : self_loop_l             -> curr (written into total at stride LDIM)
//   5    : A1^T (A_w cols 0:128)   -> u
//   6    : A2^T (A_w cols 128:256) -> v
// 8 waves share one LDS copy of M, staged async in pair-interleaved layout:
//   Bs2[q*128 + n] = { M[2q][n], M[2q+1][n] }  -> one ds_load_b64 per B frag.
// ---------------------------------------------------------------------------
__global__ __launch_bounds__(256) void node_transform_kernel(
    const float* __restrict__ x, int x_stride,
    const float* __restrict__ weights_l,
    const float* __restrict__ self_loop_l,
    const float* __restrict__ A_w_l,
    float* __restrict__ z, float* __restrict__ u, float* __restrict__ v,
    float* __restrict__ curr_out) {
  __shared__ v2f Bs2[64 * DD];  // 64 row-pairs x 128 cols = 64 KB
  const int tid = threadIdx.x;
  const int mid = blockIdx.y;

  for (int i = tid; i < DD * DD; i += 256) {
    const int k = i >> 7;
    const int n = i & (DD - 1);
    const float* g;
    if (mid < 4)        g = weights_l + mid * DD * DD + i;
    else if (mid == 4)  g = self_loop_l + i;
    else if (mid == 5)  g = A_w_l + n * ATTIN + k;
    else                g = A_w_l + n * ATTIN + DD + k;
    char* l = (char*)Bs2 + (((k >> 1) * DD + n) << 3) + ((k & 1) << 2);
    async_ld_f32(g, l);
  }
  wait_async_all();
  __syncthreads();

  const int wave = tid >> 5;
  const int lane = tid & 31;
  const int row0 = blockIdx.x * 128 + wave * 16;
  if (row0 >= NN) return;

  const int ln = lane & 15;   // A: M index / B,D: N index
  const int kh = lane >> 4;   // K half selector

  const float* xr = x + (size_t)(row0 + ln) * x_stride;

  v8f acc[8] = {};
#pragma unroll 2
  for (int k = 0; k < DD; k += 4) {
    const int q = (k >> 1) + kh;             // (k + 2*kh)/2, always integral
    const v2f a = *(const v2f*)(xr + k + 2 * kh);
#pragma unroll
    for (int t = 0; t < 8; ++t) {
      const v2f b = Bs2[q * DD + t * 16 + ln];
      acc[t] = wmma_f32(a, b, acc[t]);
    }
  }

  float* outp;
  int ostride;
  if (mid < 4)       { outp = z + (size_t)row0 * (NBASIS * DD) + mid * DD; ostride = NBASIS * DD; }
  else if (mid == 4) { outp = curr_out + (size_t)row0 * LDIM;              ostride = LDIM; }
  else if (mid == 5) { outp = u + (size_t)row0 * DD;                       ostride = DD; }
  else               { outp = v + (size_t)row0 * DD;                       ostride = DD; }

#pragma unroll
  for (int t = 0; t < 8; ++t) {
#pragma unroll
    for (int j = 0; j < 8; ++j) {
      outp[(size_t)(j + 8 * kh) * ostride + t * 16 + ln] = acc[t][j];
    }
  }
}

// ---------------------------------------------------------------------------
// Fused edge kernel: per wave, 16 edges.
//  h = relu( WMMA([rel_embed|tgt_rel] @ A34^T) + u[src] + v[tgt] + A_b )
//  a = sigmoid( h . B_w + B_b )          (butterfly shfl reduction, wave32)
//  scatter: out[tgt] += a * sum_b coef_b * z_b[src]   (atomic fp32)
// A34^T staged async into LDS, pair-interleaved like the node kernel.
// ---------------------------------------------------------------------------
__global__ __launch_bounds__(128) void edge_attn_scatter_kernel(
    const int* __restrict__ src, const int* __restrict__ tgt,
    const int* __restrict__ rel,
    const float* __restrict__ rel_embed, const float* __restrict__ tgt_rel,
    const float* __restrict__ wcomp_l,
    const float* __restrict__ u, const float* __restrict__ v,
    const float* __restrict__ z,
    const float* __restrict__ A_w_l, const float* __restrict__ A_b_l,
    const float* __restrict__ B_w_l, const float* __restrict__ B_b_l,
    float* __restrict__ out_l) {
  __shared__ v2f Ms2[32 * DD];  // 32 row-pairs x 128 cols = 32 KB
  const int tid = threadIdx.x;
  for (int i = tid; i < 64 * DD; i += 128) {
    const int k = i >> 7;
    const int n = i & (DD - 1);
    const float* g = A_w_l + n * ATTIN + 256 + k;  // M[k][n] = A_w[n, 256+k]
    char* l = (char*)Ms2 + (((k >> 1) * DD + n) << 3) + ((k & 1) << 2);
    async_ld_f32(g, l);
  }
  wait_async_all();
  __syncthreads();

  const int wave = tid >> 5;
  const int lane = tid & 31;
  const int e0 = (blockIdx.x * 4 + wave) * 16;
  if (e0 >= EE) return;

  const int ln = lane & 15;
  const int kh = lane >> 4;

  const float* rp = rel_embed + (size_t)(e0 + ln) * ARDIM;
  const float* tp = tgt_rel + (size_t)(e0 + ln) * ARDIM;

  v8f acc[8] = {};
#pragma unroll
  for (int k = 0; k < 64; k += 4) {
    const int kk = k + 2 * kh;
    const int q = (k >> 1) + kh;
    const v2f a = (kk < 32) ? *(const v2f*)(rp + kk)
                            : *(const v2f*)(tp + kk - 32);
#pragma unroll
    for (int t = 0; t < 8; ++t) {
      const v2f b = Ms2[q * DD + t * 16 + ln];
      acc[t] = wmma_f32(a, b, acc[t]);
    }
  }

  // attention scalar per edge
  float sig[8];
  const float Bb = B_b_l[0];
#pragma unroll
  for (int j = 0; j < 8; ++j) {
    const int m = j + 8 * kh;  // edge row handled by this half-wave
    const int es = src[e0 + m];
    const int et = tgt[e0 + m];
    const float* up = u + (size_t)es * DD;
    const float* vp = v + (size_t)et * DD;
    float partial = 0.0f;
#pragma unroll
    for (int t = 0; t < 8; ++t) {
      const int n = t * 16 + ln;
      float h = acc[t][j] + up[n] + vp[n] + A_b_l[n];
      h = fmaxf(h, 0.0f);
      partial += h * B_w_l[n];
    }
    // reduce across the 16 lanes of this half-wave (butterfly, all lanes get sum)
    partial += __shfl_xor(partial, 1, 32);
    partial += __shfl_xor(partial, 2, 32);
    partial += __shfl_xor(partial, 4, 32);
    partial += __shfl_xor(partial, 8, 32);
    sig[j] = 1.0f / (1.0f + __expf(-(partial + Bb)));
  }

  // scatter phase: 32 lanes x float4 cover the 128-dim message of each edge
#pragma unroll
  for (int m = 0; m < 16; ++m) {
    const float aval = __shfl(sig[m & 7], (m & 8) ? 16 : 0, 32);
    const int es = src[e0 + m];
    const int et = tgt[e0 + m];
    const int er = rel[e0 + m];
    const float4* zp = (const float4*)(z + (size_t)es * (NBASIS * DD));
    float4 s = make_float4(0.f, 0.f, 0.f, 0.f);
#pragma unroll
    for (int b = 0; b < NBASIS; ++b) {
      const float c = wcomp_l[er * NBASIS + b];
      const float4 zz = zp[b * 32 + lane];
      s.x += c * zz.x; s.y += c * zz.y; s.z += c * zz.z; s.w += c * zz.w;
    }
    float* op = out_l + (size_t)et * LDIM + lane * 4;
    atomicAdd(op + 0, s.x * aval);
    atomicAdd(op + 1, s.y * aval);
    atomicAdd(op + 2, s.z * aval);
    atomicAdd(op + 3, s.w * aval);
  }
}

__global__ __launch_bounds__(256) void relu_kernel(float* __restrict__ out_l) {
  const size_t i = (size_t)blockIdx.x * blockDim.x + threadIdx.x;
  if (i >= (size_t)NN * DD) return;
  const size_t row = i >> 7;
  const int col = (int)(i & (DD - 1));
  float* p = out_l + row * LDIM + col;
  *p = fmaxf(*p, 0.0f);
}

__global__ __launch_bounds__(256) void zero_kernel(float* __restrict__ p, int n) {
  const int i = blockIdx.x * blockDim.x + threadIdx.x;
  if (i < n) p[i] = 0.0f;
}

__global__ __launch_bounds__(256) void graph_accum_kernel(
    const float* __restrict__ total, const int* __restrict__ gid,
    float* __restrict__ gsum, float* __restrict__ gcnt) {
  const size_t i = (size_t)blockIdx.x * blockDim.x + threadIdx.x;
  if (i >= (size_t)NN * LDIM) return;
  const int node = (int)(i >> 8);
  const int c = (int)(i & (LDIM - 1));
  const int g = gid[node];
  atomicAdd(&gsum[g * LDIM + c], total[i]);
  if (c == 0) atomicAdd(&gcnt[g], 1.0f);
}

__global__ __launch_bounds__(256) void finalize_kernel(
    const float* __restrict__ total, const float* __restrict__ gsum,
    const float* __restrict__ gcnt, const int* __restrict__ srcn,
    const int* __restrict__ tgtn, float* __restrict__ out) {
  const int i = blockIdx.x * blockDim.x + threadIdx.x;
  const int GS = GG * LDIM;          // 12800
  const int SS = NSAMP * LDIM;       // 512000
  if (i < GS) {
    out[i] = gsum[i] / gcnt[i / LDIM];
  } else if (i < GS + SS) {
    const int j = i - GS;
    out[i] = total[(size_t)srcn[j / LDIM] * LDIM + (j & (LDIM - 1))];
  } else if (i < GS + 2 * SS) {
    const int j = i - GS - SS;
    out[i] = total[(size_t)tgtn[j / LDIM] * LDIM + (j & (LDIM - 1))];
  }
}

extern "C" void kernel_launch(void* const* d_in, const int* in_sizes, int n_in,
                              void* d_out, int out_size, void* d_ws,
                              size_t ws_size, hipStream_t stream) {
  const float* node_feat  = (const float*)d_in[0];
  const int*   edge       = (const int*)d_in[1];   // (2, E)
  const int*   relation   = (const int*)d_in[2];
  const float* rel_embed  = (const float*)d_in[3];
  const float* tgt_rel    = (const float*)d_in[4];
  const int*   src_node   = (const int*)d_in[5];
  const int*   tgt_node   = (const int*)d_in[6];
  const int*   graph_ids  = (const int*)d_in[7];
  const float* weights    = (const float*)d_in[8];
  const float* w_comps    = (const float*)d_in[9];
  const float* self_loops = (const float*)d_in[10];
  const float* A_w        = (const float*)d_in[11];
  const float* A_b        = (const float*)d_in[12];
  const float* B_w        = (const float*)d_in[13];
  const float* B_b        = (const float*)d_in[14];

  const int* srcp = edge;
  const int* tgtp = edge + EE;

  // workspace layout (floats)
  float* total = (float*)d_ws;                         // N * 256
  float* z     = total + (size_t)NN * LDIM;            // N * 512
  float* u     = z + (size_t)NN * NBASIS * DD;         // N * 128
  float* v     = u + (size_t)NN * DD;                  // N * 128
  float* gsum  = v + (size_t)NN * DD;                  // G * 256
  float* gcnt  = gsum + GG * LDIM;                     // G

  for (int l = 0; l < LLAY; ++l) {
    const float* xin = (l == 0) ? node_feat : total;   // layer1 reads cols 0:128
    const int xstride = (l == 0) ? DD : LDIM;
    const float* W_l  = weights + (size_t)l * NBASIS * DD * DD;
    const float* wc_l = w_comps + (size_t)l * RR * NBASIS;
    const float* sl_l = self_loops + (size_t)l * DD * DD;
    const float* Aw_l = A_w + (size_t)l * DD * ATTIN;
    const float* Ab_l = A_b + (size_t)l * DD;
    const float* Bw_l = B_w + (size_t)l * DD;
    const float* Bb_l = B_b + l;
    float* out_l = total + l * DD;

    dim3 gN((NN + 127) / 128, 7, 1);
    node_transform_kernel<<<gN, 256, 0, stream>>>(xin, xstride, W_l, sl_l, Aw_l,
                                                  z, u, v, out_l);

    dim3 gE(EE / 64, 1, 1);  // 16 edges per wave, 4 waves per block
    edge_attn_scatter_kernel<<<gE, 128, 0, stream>>>(
        srcp, tgtp, relation, rel_embed, tgt_rel, wc_l, u, v, z, Aw_l, Ab_l,
        Bw_l, Bb_l, out_l);

    dim3 gR(((size_t)NN * DD + 255) / 256, 1, 1);
    relu_kernel<<<gR, 256, 0, stream>>>(out_l);
  }

  const int nz = GG * LDIM + GG;
  zero_kernel<<<(nz + 255) / 256, 256, 0, stream>>>(gsum, nz);

  dim3 gA(((size_t)NN * LDIM + 255) / 256, 1, 1);
  graph_accum_kernel<<<gA, 256, 0, stream>>>(total, graph_ids, gsum, gcnt);

  const int nout = GG * LDIM + 2 * NSAMP * LDIM;  // 1,036,800
  finalize_kernel<<<(nout + 255) / 256, 256, 0, stream>>>(total, gsum, gcnt,
                                                          src_node, tgt_node,
                                                          (float*)d_out);
}